// SparseFlashAttn_77781857731158
// MI455X (gfx1250) — compile-verified
//
#include <hip/hip_runtime.h>
#include <math.h>

typedef float v2f __attribute__((ext_vector_type(2)));
typedef float v8f __attribute__((ext_vector_type(8)));
typedef unsigned int v4u __attribute__((ext_vector_type(4)));
typedef int v4i __attribute__((ext_vector_type(4)));
typedef int v8i __attribute__((ext_vector_type(8)));

#define BB 16
#define HH 32
#define HKVN 8
#define DD 128
#define DVV 128
#define SS 8192
#define MM 64
#define BNN 64
#define GG 4
#define NSPLIT 16
#define CHUNK 4              // KV blocks per wave
#define NSUB (CHUNK * 2)     // 32-row sub-blocks per wave
#define SROWS 32             // rows per staged sub-block
#define KSTRIDE 132          // padded LDS row stride (floats): 128 + 4 (TDM pad)
#define LDS_PAD 18           // padded row stride (floats) for P staging

// reduce over lanes 0..15 (upper half reduces its own garbage), broadcast low half
__device__ __forceinline__ float warp_max16(float v) {
  v = fmaxf(v, __shfl_xor(v, 1, 32));
  v = fmaxf(v, __shfl_xor(v, 2, 32));
  v = fmaxf(v, __shfl_xor(v, 4, 32));
  v = fmaxf(v, __shfl_xor(v, 8, 32));
  return __shfl(v, (int)(threadIdx.x & 15), 32);
}
__device__ __forceinline__ float warp_sum16(float v) {
  v += __shfl_xor(v, 1, 32);
  v += __shfl_xor(v, 2, 32);
  v += __shfl_xor(v, 4, 32);
  v += __shfl_xor(v, 8, 32);
  return __shfl(v, (int)(threadIdx.x & 15), 32);
}

// TDM: DMA a [rows x 128] f32 tile (row stride 1024 floats in global) into LDS,
// padding 4 DWORDs after every 128 DWORDs -> LDS row stride = 132 floats.
// D# layout per CDNA5 ISA.
// Toolchain uses the 6-arg builtin: (v4u g0, v8i g1, v4i g2, v4i g3, v8i, i32 cpol).
__device__ __forceinline__ void tdm_load_2d(unsigned lds_off, const float* gsrc,
                                            unsigned rows) {
  unsigned long long ga = (unsigned long long)(uintptr_t)gsrc;
  v4u g0;
  g0[0] = 1u;                                   // count=1 valid user descriptor
  g0[1] = lds_off;                              // lds_addr (bytes)
  g0[2] = (unsigned)(ga & 0xFFFFFFFFu);         // global_addr[31:0]
  g0[3] = (unsigned)((ga >> 32) & 0x01FFFFFFu)  // global_addr[56:32]
          | (2u << 30);                         // type=2 ("image")
  v8i g1;
  g1[0] = (int)((2u << 16)      // data_size = 4 bytes
                | (1u << 20)    // pad_enable
                | (6u << 22)    // pad_interval: every 128 DWORDs
                | (3u << 25));  // pad_amount: 4 DWORDs
  g1[1] = (int)(128u << 16);                  // tensor_dim0 = 128 (low 16)
  g1[2] = (int)(rows << 16);                  // tensor_dim0 hi | tensor_dim1 lo
  g1[3] = (int)(128u << 16);                  // tensor_dim1 hi | tile_dim0 = 128
  g1[4] = (int)rows;                          // tile_dim1 | tile_dim2 = 0
  g1[5] = 1024;                               // tensor_dim0_stride = HKV*D
  g1[6] = 0;
  g1[7] = 0;
  v4i z4 = {0, 0, 0, 0};
  v8i z8 = {0, 0, 0, 0, 0, 0, 0, 0};
  __builtin_amdgcn_tensor_load_to_lds(g0, g1, z4, z4, z8, 0);
}

// Kernel 1: one wave per (b, hkv, split). TDM double-buffered K/V staging into
// LDS, fp32 WMMA for QK^T and P*V, online softmax. Writes unnormalized partials.
__global__ void __launch_bounds__(32)
sfa_partial(const float* __restrict__ q,
            const float* __restrict__ K,
            const float* __restrict__ V,
            const int* __restrict__ bidx,
            const int* __restrict__ seqlens,
            float* __restrict__ o_part,    // [B,H,NSPLIT,DV]
            float* __restrict__ ml_part)   // [B,H,NSPLIT,2]
{
  __shared__ float kbuf[2][SROWS * KSTRIDE];
  __shared__ float vbuf[2][SROWS * KSTRIDE];
  __shared__ float pbuf[16 * LDS_PAD];

  const int wg    = blockIdx.x;
  const int split = wg % NSPLIT;
  const int hkv   = (wg / NSPLIT) % HKVN;
  const int b     = wg / (NSPLIT * HKVN);
  const int lane  = threadIdx.x;
  const int col   = lane & 15;    // N index within a 16-tile / A-matrix row
  const int kh    = lane >> 4;    // K-half select for A/B fragments

  const unsigned koff[2] = {(unsigned)(uintptr_t)&kbuf[0][0],
                            (unsigned)(uintptr_t)&kbuf[1][0]};
  const unsigned voff[2] = {(unsigned)(uintptr_t)&vbuf[0][0],
                            (unsigned)(uintptr_t)&vbuf[1][0]};

  // zero P staging (rows 4..7, 12..15 must stay zero forever)
  for (int i = lane; i < 16 * LDS_PAD; i += 32) pbuf[i] = 0.0f;
  asm volatile("s_wait_dscnt 0" ::: "memory");

  const float scale  = 0.08838834764831845f; // 1/sqrt(128)
  const int   seqlen = seqlens[b];

  const float* Kb = K + (((size_t)b * SS) * HKVN + hkv) * DD;
  const float* Vb = V + (((size_t)b * SS) * HKVN + hkv) * DVV;
  const int*   bi = bidx + ((size_t)b * HKVN + hkv) * MM + split * CHUNK;

  // ---- prologue: DMA sub-block 0 into buffer 0 ----
  {
    const int blk0 = bi[0];
    const int r0 = ((blk0 >= 0) ? blk0 : 0) * BNN;
    tdm_load_2d(koff[0], Kb + (size_t)r0 * (HKVN * DD), SROWS);
    tdm_load_2d(voff[0], Vb + (size_t)r0 * (HKVN * DVV), SROWS);
  }

  // ---- Q as A-fragments: A(m,k): lane = m + 16*(k>>1), vgpr = k&1 ----
  const int   qrow   = col;
  const float rowsel = (qrow < GG) ? scale : 0.0f;  // zero-pad rows 4..15, fold scale
  const float* qp = q + ((size_t)b * HH + hkv * GG + (qrow < GG ? qrow : 0)) * DD;
  v2f aq[32];
#pragma unroll
  for (int t = 0; t < 32; ++t) {
    const int k0 = 4 * t + 2 * kh;
    v2f qv = *(const v2f*)(qp + k0);
    aq[t][0] = qv[0] * rowsel;
    aq[t][1] = qv[1] * rowsel;
  }

  float mrow[GG], lrow[GG];
#pragma unroll
  for (int g = 0; g < GG; ++g) { mrow[g] = -3.0e38f; lrow[g] = 0.0f; }

  v8f oacc[8];
#pragma unroll
  for (int ot = 0; ot < 8; ++ot)
#pragma unroll
    for (int v = 0; v < 8; ++v) oacc[ot][v] = 0.0f;

  for (int s = 0; s < NSUB; ++s) {
    // our LDS reads of the buffer about to be overwritten must retire first
    asm volatile("s_wait_dscnt 0" ::: "memory");
    if (s + 1 < NSUB) {
      const int nblk = bi[(s + 1) >> 1];
      const int nr0  = ((nblk >= 0) ? nblk : 0) * BNN + ((s + 1) & 1) * SROWS;
      tdm_load_2d(koff[(s + 1) & 1], Kb + (size_t)nr0 * (HKVN * DD), SROWS);
      tdm_load_2d(voff[(s + 1) & 1], Vb + (size_t)nr0 * (HKVN * DVV), SROWS);
      __builtin_amdgcn_s_wait_tensorcnt(2);   // sub-block s landed (TDM in-order)
    } else {
      __builtin_amdgcn_s_wait_tensorcnt(0);
    }
    asm volatile("" ::: "memory");

    const int blk = bi[s >> 1];               // wave-uniform
    if (blk < 0) continue;
    const int base = blk * BNN + (s & 1) * SROWS;
    const float* kb = &kbuf[s & 1][0];
    const float* vb = &vbuf[s & 1][0];

    // ---- S = (Q*scale) . K^T : two 16-col tiles, 32 wmma k-steps each ----
    v8f sc[2];
#pragma unroll
    for (int nt = 0; nt < 2; ++nt)
#pragma unroll
      for (int v = 0; v < 8; ++v) sc[nt][v] = 0.0f;
#pragma unroll
    for (int nt = 0; nt < 2; ++nt) {
      const float* Krow = kb + (size_t)(nt * 16 + col) * KSTRIDE;
#pragma unroll
      for (int t = 0; t < 32; ++t) {
        // B(k,n): lane = n + 16*(k>>1), vgpr = k&1 ; B[k][n] = K[pos=n][4t+k]
        v2f bf = *(const v2f*)(Krow + 4 * t + 2 * kh);
        sc[nt] = __builtin_amdgcn_wmma_f32_16x16x4_f32(
            false, aq[t], false, bf, (short)0, sc[nt], false, false);
      }
    }

    // ---- per-row max over this sub-block ----
    const int pos0 = base + col;
    float mnew[GG], alpha[GG];
#pragma unroll
    for (int g = 0; g < GG; ++g) {
      float mx = -3.0e38f;
#pragma unroll
      for (int nt = 0; nt < 2; ++nt) {
        const bool valid = (pos0 + nt * 16) < seqlen;
        mx = fmaxf(mx, valid ? sc[nt][g] : -3.0e38f);
      }
      mx = warp_max16(mx);
      mnew[g]  = fmaxf(mrow[g], mx);
      alpha[g] = __expf(mrow[g] - mnew[g]);
      mrow[g]  = mnew[g];
    }

    // rescale accumulator (pad rows are zero -> any factor is safe)
#pragma unroll
    for (int ot = 0; ot < 8; ++ot)
#pragma unroll
      for (int v = 0; v < 8; ++v)
        oacc[ot][v] *= (v < GG) ? alpha[v] : 1.0f;

    // ---- P tiles: exp, LDS transpose (C-layout -> A-layout), P*V wmma ----
    float rs[GG] = {0.0f, 0.0f, 0.0f, 0.0f};
#pragma unroll
    for (int nt = 0; nt < 2; ++nt) {
      const bool valid = (pos0 + nt * 16) < seqlen;
#pragma unroll
      for (int v = 0; v < GG; ++v) {
        // lanes 0..15 write rows 0..3 (exp), lanes 16..31 write zeros to rows 8..11
        float p = (valid && kh == 0) ? __expf(sc[nt][v] - mnew[v]) : 0.0f;
        rs[v] += p;
        pbuf[(v + kh * 8) * LDS_PAD + col] = p;
      }
      asm volatile("s_wait_dscnt 0" ::: "memory");
      // A(m=g, k=n-in-tile): lane m = col, k = 4*t2 + 2*kh + {0,1}
      v2f ap[4];
#pragma unroll
      for (int t2 = 0; t2 < 4; ++t2)
        ap[t2] = *(const v2f*)(pbuf + col * LDS_PAD + 4 * t2 + 2 * kh);

#pragma unroll
      for (int t2 = 0; t2 < 4; ++t2) {
        const float* Vk = vb + (size_t)(nt * 16 + 4 * t2 + 2 * kh) * KSTRIDE;
#pragma unroll
        for (int ot = 0; ot < 8; ++ot) {
          // B(k,dv): lane dv = ot*16+col, k = 2*kh + vgpr -> consecutive V rows
          v2f bv;
          bv[0] = Vk[ot * 16 + col];
          bv[1] = Vk[KSTRIDE + ot * 16 + col];
          oacc[ot] = __builtin_amdgcn_wmma_f32_16x16x4_f32(
              false, ap[t2], false, bv, (short)0, oacc[ot], false, false);
        }
      }
    }

#pragma unroll
    for (int g = 0; g < GG; ++g)
      lrow[g] = lrow[g] * alpha[g] + warp_sum16(rs[g]);
  }

  // ---- write partials: o rows 0..3 (vgpr 0..3 of lanes 0..15), plus (m,l) ----
  if (kh == 0) {
#pragma unroll
    for (int v = 0; v < GG; ++v) {
      float* oph = o_part + (((size_t)b * HH + hkv * GG + v) * NSPLIT + split) * DVV;
#pragma unroll
      for (int ot = 0; ot < 8; ++ot)
        oph[ot * 16 + col] = oacc[ot][v];
    }
  }
  if (lane == 0) {
#pragma unroll
    for (int g = 0; g < GG; ++g) {
      float* mlp = ml_part + (((size_t)b * HH + hkv * GG + g) * NSPLIT + split) * 2;
      mlp[0] = mrow[g];
      mlp[1] = lrow[g];
    }
  }
}

// Kernel 2: log-sum-exp combine of NSPLIT partials per (b,h); one thread per dv.
__global__ void __launch_bounds__(128)
sfa_combine(const float* __restrict__ o_part,
            const float* __restrict__ ml_part,
            float* __restrict__ out)
{
  const int bh = blockIdx.x;       // b*H + h
  const int dv = threadIdx.x;      // 0..127
  const float* ml = ml_part + (size_t)bh * NSPLIT * 2;
  float mmax = -3.0e38f;
#pragma unroll
  for (int s = 0; s < NSPLIT; ++s) mmax = fmaxf(mmax, ml[2 * s]);
  float ltot = 0.0f, acc = 0.0f;
  const float* op = o_part + (size_t)bh * NSPLIT * DVV;
#pragma unroll
  for (int s = 0; s < NSPLIT; ++s) {
    const float w = __expf(ml[2 * s] - mmax);
    ltot += ml[2 * s + 1] * w;
    acc  += op[(size_t)s * DVV + dv] * w;
  }
  const float inv = (ltot > 0.0f) ? (1.0f / ltot) : 0.0f;
  out[(size_t)bh * DVV + dv] = acc * inv;
}

extern "C" void kernel_launch(void* const* d_in, const int* in_sizes, int n_in,
                              void* d_out, int out_size, void* d_ws, size_t ws_size,
                              hipStream_t stream) {
  const float* q    = (const float*)d_in[0];
  const float* K    = (const float*)d_in[1];
  const float* V    = (const float*)d_in[2];
  const int*   bidx = (const int*)d_in[3];
  const int*   seql = (const int*)d_in[4];
  // d_in[5] = block_size scalar; BN is baked in at compile time.

  float* o_part  = (float*)d_ws;                               // B*H*NSPLIT*DV floats
  float* ml_part = o_part + (size_t)BB * HH * NSPLIT * DVV;    // B*H*NSPLIT*2 floats

  sfa_partial<<<dim3(BB * HKVN * NSPLIT), dim3(32), 0, stream>>>(
      q, K, V, bidx, seql, o_part, ml_part);
  sfa_combine<<<dim3(BB * HH), dim3(128), 0, stream>>>(
      o_part, ml_part, (float*)d_out);
}